// HGTLayer_61005715472534
// MI455X (gfx1250) — compile-verified
//
#include <hip/hip_runtime.h>
#include <stdint.h>

static constexpr int DIM = 256;
static constexpr int HH  = 8;
static constexpr int NT  = 6;
static constexpr int ET  = 3;

typedef float v2f __attribute__((ext_vector_type(2)));
typedef float v8f __attribute__((ext_vector_type(8)));

// D = A(16x4,f32) * B(4x16,f32) + C(16x16,f32)
#define WMMA4(a, b, c) \
  __builtin_amdgcn_wmma_f32_16x16x4_f32(false, (a), false, (b), (short)0, (c), false, false)

// ---------------- Tensor Data Mover (TDM) helpers ----------------
#if defined(__has_builtin)
#if __has_builtin(__builtin_amdgcn_tensor_load_to_lds)
#define HAVE_TDM 1
#endif
#endif
#ifndef HAVE_TDM
#define HAVE_TDM 0
#endif

#if HAVE_TDM
typedef unsigned int tdm_v4u __attribute__((ext_vector_type(4)));
typedef int          tdm_v4i __attribute__((ext_vector_type(4)));
typedef int          tdm_v8i __attribute__((ext_vector_type(8)));

__device__ __forceinline__ void tdm_call(tdm_v4u g0, tdm_v8i g1, tdm_v4i g2, tdm_v4i g3) {
#if __has_include(<hip/amd_detail/amd_gfx1250_TDM.h>)
  tdm_v8i z8 = {0, 0, 0, 0, 0, 0, 0, 0};
  __builtin_amdgcn_tensor_load_to_lds(g0, g1, g2, g3, z8, 0);
#else
  __builtin_amdgcn_tensor_load_to_lds(g0, g1, g2, g3, 0);
#endif
}

// D# group1 for a 256-wide f32 tile, 16 rows, row-stride 256 elems,
// LDS padding: +1 DWORD every 256 DWORDs -> LDS row stride 257 floats.
__device__ __forceinline__ tdm_v8i tdm_group1(unsigned tensorRows) {
  tdm_v8i g1;
  g1[0] = (2 << 16) | (1 << 20) | (7 << 22);          // data_size=4B | pad_enable | pad_interval=256DW (amount=1DW)
  g1[1] = (int)(256u << 16);                          // tensor_dim0[15:0]=256 in bits[63:48]
  g1[2] = (int)((tensorRows & 0xFFFFu) << 16);        // tensor_dim0 hi=0 | tensor_dim1 lo
  g1[3] = (int)(((tensorRows >> 16) & 0xFFFFu) | (256u << 16)); // tensor_dim1 hi | tile_dim0=256
  g1[4] = 16;                                         // tile_dim1=16, tile_dim2=0
  g1[5] = 256;                                        // tensor_dim0_stride lo = 256 elems
  g1[6] = 0;                                          // stride hi | dim1_stride lo
  g1[7] = 0;
  return g1;
}

// Dense 2D tile load: 16 consecutive rows starting at g (rows clamped by tensorRows -> OOB reads 0)
__device__ __forceinline__ void tdm_load_tile16(const float* g, unsigned ldsOff, unsigned tensorRows) {
  unsigned long long ga = (unsigned long long)(uintptr_t)g;
  tdm_v4u g0;
  g0.x = 1u;                                          // count=1, no gather
  g0.y = ldsOff;                                      // LDS byte address
  g0.z = (unsigned)ga;
  g0.w = (unsigned)((ga >> 32) & 0x01FFFFFFu) | (2u << 30);  // global_addr[56:32] | type=2
  tdm_v4i z4 = {0, 0, 0, 0};
  tdm_call(g0, tdm_group1(tensorRows), z4, z4);
}

// Gather-mode load of 16 rows (16-bit indices; N < 65536 here)
__device__ __forceinline__ void tdm_gather_tile16(const float* g, unsigned ldsOff,
                                                  const int* rows16, unsigned tensorRows) {
  unsigned long long ga = (unsigned long long)(uintptr_t)g;
  tdm_v4u g0;
  g0.x = 1u | (1u << 31);                             // count=1 | gather_mode (16-bit indices)
  g0.y = ldsOff;
  g0.z = (unsigned)ga;
  g0.w = (unsigned)((ga >> 32) & 0x01FFFFFFu) | (2u << 30);
  tdm_v4i g2, g3;
  #pragma unroll
  for (int j = 0; j < 4; ++j) {
    g2[j] = (rows16[2 * j] & 0xFFFF) | ((rows16[2 * j + 1] & 0xFFFF) << 16);
    g3[j] = (rows16[8 + 2 * j] & 0xFFFF) | ((rows16[8 + 2 * j + 1] & 0xFFFF) << 16);
  }
  tdm_call(g0, tdm_group1(tensorRows), g2, g3);
}
#endif  // HAVE_TDM

__device__ __forceinline__ void atomicMaxF(float* addr, float v) {
  if (v >= 0.0f) atomicMax((int*)addr, __float_as_int(v));
  else           atomicMin((unsigned int*)addr, __float_as_uint(v));
}

// ---------------- control / init ----------------
// ctrl layout: [0..5]=typeCount [8..14]=typeOffset [16..21]=cursor [24..30]=tileStart
__global__ void k_init_ctrl(int* ctrl, float* rhMax, float* rhSum) {
  int t = threadIdx.x;
  if (t < 64) ctrl[t] = 0;
  if (t < 32) { rhMax[t] = -__builtin_inff(); rhSum[t] = 0.0f; }
}

__global__ void k_zero(float* p, long n) {
  long i = (long)blockIdx.x * blockDim.x + threadIdx.x;
  long stride = (long)gridDim.x * blockDim.x;
  for (; i < n; i += stride) p[i] = 0.0f;
}

__global__ void k_count(const int* __restrict__ ntype, int* ctrl, int N) {
  int i = blockIdx.x * blockDim.x + threadIdx.x;
  if (i < N) atomicAdd(&ctrl[ntype[i]], 1);
}

__global__ void k_scan(int* ctrl) {
  int off = 0, toff = 0;
  for (int t = 0; t < NT; ++t) {
    ctrl[8 + t] = off;
    ctrl[24 + t] = toff;
    off += ctrl[t];
    toff += (ctrl[t] + 15) >> 4;
  }
  ctrl[8 + NT] = off;
  ctrl[24 + NT] = toff;
}

__global__ void k_scatter(const int* __restrict__ ntype, int* ctrl, int* perm, int N) {
  int i = blockIdx.x * blockDim.x + threadIdx.x;
  if (i < N) {
    int ty = ntype[i];
    int pos = atomicAdd(&ctrl[16 + ty], 1);
    perm[ctrl[8 + ty] + pos] = i;
  }
}

// ---------------- type-bucketed K/Q/V projection (WMMA f32 16x16x4) ----------------
// One block = 16 nodes of one type. K/Q/V at the same col0 share one A fragment:
// 3 independent accumulators break the WMMA->WMMA dependency chain.
__global__ __launch_bounds__(128) void k_proj(
    const float* __restrict__ x, const int* __restrict__ perm, const int* __restrict__ ctrl,
    const float* __restrict__ Wk, const float* __restrict__ bk,
    const float* __restrict__ Wq, const float* __restrict__ bq,
    const float* __restrict__ Wv, const float* __restrict__ bv,
    float* __restrict__ Kb, float* __restrict__ Qb, float* __restrict__ Vb) {
  __shared__ float As[16][257];
  __shared__ int s_node[16];
  int b = blockIdx.x;
  if (b >= ctrl[24 + NT]) return;
  int ty = 0;
  for (int t = 0; t < NT; ++t)
    if (b >= ctrl[24 + t]) ty = t;
  int localTile = b - ctrl[24 + ty];
  int count = ctrl[ty];
  int base = ctrl[8 + ty] + localTile * 16;
  int nvalid = count - localTile * 16; if (nvalid > 16) nvalid = 16;
  int tid = threadIdx.x;
  if (tid < 16) {
    int idx = (tid < nvalid) ? (base + tid) : base;  // pad with a valid row
    s_node[tid] = perm[idx];
  }
  __syncthreads();
#if HAVE_TDM
  if (tid < 32) {
    int rows[16];
    #pragma unroll
    for (int i = 0; i < 16; ++i) rows[i] = s_node[i];
    tdm_gather_tile16(x, (unsigned)(uintptr_t)&As[0][0], rows, 0x7FFFFFFFu);
    __builtin_amdgcn_s_wait_tensorcnt(0);
  }
#else
  #pragma unroll
  for (int i = 0; i < 32; ++i) {
    int lin = i * 128 + tid;
    int row = lin >> 8, col = lin & 255;
    As[row][col] = x[(long)s_node[row] * DIM + col];
  }
#endif
  __syncthreads();

  int lane = tid & 31, laneLo = lane & 15, hiHalf = lane >> 4, wv = tid >> 5;
  const long tyW = (long)ty * DIM * DIM;
  for (int ct = 0; ct < 4; ++ct) {
    int col0 = (wv * 4 + ct) * 16;
    const float* WtK = Wk + tyW + col0 + laneLo;
    const float* WtQ = Wq + tyW + col0 + laneLo;
    const float* WtV = Wv + tyW + col0 + laneLo;
    v8f cK = {}, cQ = {}, cV = {};
    for (int kk = 0; kk < 64; ++kk) {               // K = 256 in steps of 4
      int kb = kk * 4 + hiHalf * 2;
      v2f a, bK, bQ, bV;
      a.x = As[laneLo][kb];
      a.y = As[laneLo][kb + 1];
      long o0 = (long)kb * DIM, o1 = (long)(kb + 1) * DIM;
      bK.x = WtK[o0]; bK.y = WtK[o1];
      bQ.x = WtQ[o0]; bQ.y = WtQ[o1];
      bV.x = WtV[o0]; bV.y = WtV[o1];
      cK = WMMA4(a, bK, cK);
      cQ = WMMA4(a, bQ, cQ);
      cV = WMMA4(a, bV, cV);
    }
    float bvK = bk[ty * DIM + col0 + laneLo];
    float bvQ = bq[ty * DIM + col0 + laneLo];
    float bvV = bv[ty * DIM + col0 + laneLo];
    #pragma unroll
    for (int j = 0; j < 8; ++j) {
      int row = j + hiHalf * 8;
      if (row < nvalid) {
        long o = (long)s_node[row] * DIM + col0 + laneLo;
        Kb[o] = cK[j] + bvK;
        Qb[o] = cQ[j] + bvQ;
        Vb[o] = cV[j] + bvV;
      }
    }
  }
}

// ---------------- Qhat[n,r,h,d] = sum_k Wa[r][d,k]*Q[n,h,k] ; Vhat[n,r,h,k] = sum_d V[n,h,d]*Wm[r][d,k]
// Both 16-col halves share one A fragment -> 2 accumulators per (mat,r,h).
__global__ __launch_bounds__(128) void k_relhat(
    const float* __restrict__ Qb, const float* __restrict__ Vb,
    const float* __restrict__ Wa, const float* __restrict__ Wm,
    float* __restrict__ Qhat, float* __restrict__ Vhat, int N) {
  __shared__ float Qs[16][257];
  __shared__ float Vs[16][257];
  int n0 = blockIdx.x * 16;
  int tid = threadIdx.x;
#if HAVE_TDM
  if (tid < 32) {
    unsigned remRows = (unsigned)(N - n0);           // OOB rows read as zero
    tdm_load_tile16(Qb + (long)n0 * DIM, (unsigned)(uintptr_t)&Qs[0][0], remRows);
    tdm_load_tile16(Vb + (long)n0 * DIM, (unsigned)(uintptr_t)&Vs[0][0], remRows);
    __builtin_amdgcn_s_wait_tensorcnt(0);
  }
#else
  #pragma unroll
  for (int i = 0; i < 32; ++i) {
    int lin = i * 128 + tid;
    int row = lin >> 8, col = lin & 255;
    int n = n0 + row; if (n >= N) n = N - 1;
    Qs[row][col] = Qb[(long)n * DIM + col];
    Vs[row][col] = Vb[(long)n * DIM + col];
  }
#endif
  __syncthreads();
  int lane = tid & 31, laneLo = lane & 15, hiHalf = lane >> 4, wv = tid >> 5;
  for (int it = 0; it < 12; ++it) {
    int t3 = wv * 12 + it;                  // 0..47 = mat(2) x r(3) x h(8)
    int mat = t3 / 24;
    int rem = t3 % 24;
    int r = rem >> 3;
    int h = rem & 7;
    v8f c0 = {}, c1 = {};
    for (int kk = 0; kk < 8; ++kk) {        // K = 32 in steps of 4
      int kb = kk * 4 + hiHalf * 2;
      v2f a, b0, b1;
      if (mat == 0) {
        a.x = Qs[laneLo][h * 32 + kb];
        a.y = Qs[laneLo][h * 32 + kb + 1];
        b0.x = Wa[r * 1024 + laneLo * 32 + kb];            // B[k][d] = Wa[d,k]
        b0.y = Wa[r * 1024 + laneLo * 32 + kb + 1];
        b1.x = Wa[r * 1024 + (16 + laneLo) * 32 + kb];
        b1.y = Wa[r * 1024 + (16 + laneLo) * 32 + kb + 1];
      } else {
        a.x = Vs[laneLo][h * 32 + kb];
        a.y = Vs[laneLo][h * 32 + kb + 1];
        b0.x = Wm[r * 1024 + kb * 32 + laneLo];            // B[d][k] = Wm[d,k]
        b0.y = Wm[r * 1024 + (kb + 1) * 32 + laneLo];
        b1.x = Wm[r * 1024 + kb * 32 + 16 + laneLo];
        b1.y = Wm[r * 1024 + (kb + 1) * 32 + 16 + laneLo];
      }
      c0 = WMMA4(a, b0, c0);
      c1 = WMMA4(a, b1, c1);
    }
    float* O = (mat == 0) ? Qhat : Vhat;
    #pragma unroll
    for (int j = 0; j < 8; ++j) {
      int row = j + hiHalf * 8;
      int n = n0 + row;
      if (n < N) {
        long o = (long)n * (ET * DIM) + r * DIM + h * 32 + laneLo;
        O[o] = c0[j];
        O[o + 16] = c1[j];
      }
    }
  }
}

// ---------------- edge scores + per-(type,head) max ----------------
__global__ void k_scores(
    const float* __restrict__ Kb, const float* __restrict__ Qhat,
    const int* __restrict__ src, const int* __restrict__ tgt,
    const int* __restrict__ etype, const float* __restrict__ pri,
    float* __restrict__ scores, float* __restrict__ rhMax, int E) {
  __shared__ float smax[ET * HH];
  int tid = threadIdx.x;
  if (tid < ET * HH) smax[tid] = -__builtin_inff();
  __syncthreads();
  long idx = (long)blockIdx.x * blockDim.x + tid;
  if (idx < (long)E * HH) {
    int e = (int)(idx >> 3), h = (int)(idx & 7);
    int s = src[e], t = tgt[e], r = etype[e];
    int rh = r * HH + h;
    const float4* kp = (const float4*)(Kb + (long)s * DIM + h * 32);
    const float4* qp = (const float4*)(Qhat + (long)t * (ET * DIM) + r * DIM + h * 32);
    float acc = 0.0f;
    #pragma unroll
    for (int i = 0; i < 8; ++i) {
      float4 a = kp[i], b = qp[i];
      acc += a.x * b.x + a.y * b.y + a.z * b.z + a.w * b.w;
    }
    acc *= 0.17677669529663687f * pri[rh];  // 1/sqrt(32)
    scores[idx] = acc;
    atomicMaxF(&smax[rh], acc);
  }
  __syncthreads();
  if (tid < ET * HH) atomicMaxF(&rhMax[tid], smax[tid]);
}

// ---------------- per-(type,head) sum of exp ----------------
__global__ void k_sumexp(const float* __restrict__ scores, const int* __restrict__ etype,
                         const float* __restrict__ rhMax, float* __restrict__ rhSum, int E) {
  __shared__ float ssum[ET * HH];
  int tid = threadIdx.x;
  if (tid < ET * HH) ssum[tid] = 0.0f;
  __syncthreads();
  long idx = (long)blockIdx.x * blockDim.x + tid;
  if (idx < (long)E * HH) {
    int e = (int)(idx >> 3), h = (int)(idx & 7);
    int rh = etype[e] * HH + h;
    atomicAdd(&ssum[rh], __expf(scores[idx] - rhMax[rh]));
  }
  __syncthreads();
  if (tid < ET * HH) atomicAdd(&rhSum[tid], ssum[tid]);
}

// ---------------- weighted message scatter-add (agg stays in 192MB L2) ----------------
__global__ void k_message(const float* __restrict__ scores, const float* __restrict__ Vhat,
                          const int* __restrict__ src, const int* __restrict__ tgt,
                          const int* __restrict__ etype,
                          const float* __restrict__ rhMax, const float* __restrict__ rhSum,
                          float* __restrict__ agg, int E) {
  long idx = (long)blockIdx.x * blockDim.x + threadIdx.x;
  if (idx >= (long)E * HH) return;
  int e = (int)(idx >> 3), h = (int)(idx & 7);
  int s = src[e], t = tgt[e], r = etype[e];
  int rh = r * HH + h;
  float denom = rhSum[rh];
  float w = (denom > 0.0f) ? __expf(scores[idx] - rhMax[rh]) / denom : 0.0f;
  const float4* vp = (const float4*)(Vhat + (long)s * (ET * DIM) + r * DIM + h * 32);
  float* ap = agg + (long)t * DIM + h * 32;
  #pragma unroll
  for (int i = 0; i < 8; ++i) {
    float4 v = vp[i];
    atomicAdd(ap + i * 4 + 0, w * v.x);
    atomicAdd(ap + i * 4 + 1, w * v.y);
    atomicAdd(ap + i * 4 + 2, w * v.z);
    atomicAdd(ap + i * 4 + 3, w * v.w);
  }
}

// ---------------- skip + LayerNorm (one wave32 per node) ----------------
__global__ void k_layernorm(const float* __restrict__ agg, const float* __restrict__ x,
                            const float* __restrict__ g, const float* __restrict__ bta,
                            float* __restrict__ out, int N) {
  int lane = threadIdx.x & 31;
  int node = blockIdx.x * 8 + (threadIdx.x >> 5);
  if (node >= N) return;
  float v[8];
  float sum = 0.0f, sq = 0.0f;
  #pragma unroll
  for (int i = 0; i < 8; ++i) {
    int col = i * 32 + lane;
    float t = agg[(long)node * DIM + col] + x[(long)node * DIM + col];
    v[i] = t; sum += t; sq += t * t;
  }
  #pragma unroll
  for (int off = 16; off > 0; off >>= 1) {
    sum += __shfl_xor(sum, off, 32);
    sq  += __shfl_xor(sq,  off, 32);
  }
  float mu  = sum * (1.0f / DIM);
  float var = sq * (1.0f / DIM) - mu * mu;
  float inv = rsqrtf(var + 1e-5f);
  #pragma unroll
  for (int i = 0; i < 8; ++i) {
    int col = i * 32 + lane;
    out[(long)node * DIM + col] = (v[i] - mu) * inv * g[col] + bta[col];
  }
}

extern "C" void kernel_launch(void* const* d_in, const int* in_sizes, int n_in,
                              void* d_out, int out_size, void* d_ws, size_t ws_size,
                              hipStream_t stream) {
  const float* x    = (const float*)d_in[0];
  const int*   eidx = (const int*)d_in[1];
  const int*   ntype= (const int*)d_in[3];
  const int*   etype= (const int*)d_in[4];
  const float* Wk   = (const float*)d_in[5];
  const float* bk   = (const float*)d_in[6];
  const float* Wq   = (const float*)d_in[7];
  const float* bq   = (const float*)d_in[8];
  const float* Wv   = (const float*)d_in[9];
  const float* bv   = (const float*)d_in[10];
  const float* pri  = (const float*)d_in[11];
  const float* Wa   = (const float*)d_in[12];
  const float* Wm   = (const float*)d_in[13];
  const float* lg   = (const float*)d_in[14];
  const float* lb   = (const float*)d_in[15];
  float* out = (float*)d_out;
  (void)n_in; (void)out_size; (void)ws_size;

  int N = in_sizes[3];
  int E = in_sizes[4];
  const int* src = eidx;
  const int* tgt = eidx + E;

  char* p = (char*)d_ws;
  auto carve = [&](size_t bytes) { char* r = p; p += (bytes + 255) & ~(size_t)255; return r; };
  int*   ctrl   = (int*)  carve(64 * sizeof(int));
  float* stats  = (float*)carve(64 * sizeof(float));
  float* rhMax  = stats;
  float* rhSum  = stats + 32;
  int*   perm   = (int*)  carve((size_t)N * sizeof(int));
  float* Kb     = (float*)carve((size_t)N * DIM * sizeof(float));
  float* Qb     = (float*)carve((size_t)N * DIM * sizeof(float));
  float* Vb     = (float*)carve((size_t)N * DIM * sizeof(float));
  float* Qhat   = (float*)carve((size_t)N * ET * DIM * sizeof(float));
  float* Vhat   = (float*)carve((size_t)N * ET * DIM * sizeof(float));
  float* scores = (float*)carve((size_t)E * HH * sizeof(float));
  float* agg    = (float*)carve((size_t)N * DIM * sizeof(float));

  k_init_ctrl<<<1, 64, 0, stream>>>(ctrl, rhMax, rhSum);
  k_zero<<<2048, 256, 0, stream>>>(agg, (long)N * DIM);
  k_count<<<(N + 255) / 256, 256, 0, stream>>>(ntype, ctrl, N);
  k_scan<<<1, 1, 0, stream>>>(ctrl);
  k_scatter<<<(N + 255) / 256, 256, 0, stream>>>(ntype, ctrl, perm, N);

  int maxTiles = (N + 15) / 16 + NT;   // upper bound incl. per-type padding
  k_proj<<<maxTiles, 128, 0, stream>>>(x, perm, ctrl, Wk, bk, Wq, bq, Wv, bv, Kb, Qb, Vb);
  k_relhat<<<(N + 15) / 16, 128, 0, stream>>>(Qb, Vb, Wa, Wm, Qhat, Vhat, N);

  long EH = (long)E * HH;
  int ehBlocks = (int)((EH + 255) / 256);
  k_scores<<<ehBlocks, 256, 0, stream>>>(Kb, Qhat, src, tgt, etype, pri, scores, rhMax, E);
  k_sumexp<<<ehBlocks, 256, 0, stream>>>(scores, etype, rhMax, rhSum, E);
  k_message<<<ehBlocks, 256, 0, stream>>>(scores, Vhat, src, tgt, etype, rhMax, rhSum, agg, E);
  k_layernorm<<<(N + 7) / 8, 256, 0, stream>>>(agg, x, lg, lb, out, N);
}